// NeuralCDE_9887014715890
// MI455X (gfx1250) — compile-verified
//
#include <hip/hip_runtime.h>
#include <math.h>

#define BB 256
#define TT 512
#define DD 64
#define HH 128
#define WW 256
#define HD (HH*DD)

typedef float v2f __attribute__((ext_vector_type(2)));
typedef float v8f __attribute__((ext_vector_type(8)));
typedef int   v4i __attribute__((ext_vector_type(4)));

static __device__ __forceinline__ v8f wmma4(v2f a, v2f b, v8f c) {
  // D = A(16x4,f32) * B(4x16,f32) + C(16x16,f32)
  return __builtin_amdgcn_wmma_f32_16x16x4_f32(false, a, false, b, (short)0, c, false, false);
}

// gfx1250 hardware tanh when the builtin exists; cheap exp-based fallback.
static __device__ __forceinline__ float fast_tanh(float x) {
#if __has_builtin(__builtin_amdgcn_tanhf)
  return __builtin_amdgcn_tanhf(x);
#else
  float e = __expf(2.0f * x);
  return 1.0f - 2.0f / (e + 1.0f);
#endif
}

static __device__ __forceinline__ float fast_softplus(float x) {
  return (x > 20.0f) ? x : __logf(1.0f + __expf(x));
}

// Async memory->LDS copy of 16 bytes per lane (CDNA5 GLOBAL_LOAD_ASYNC_TO_LDS_B128),
// falling back to a synchronous VGPR round-trip when the builtin is absent.
#if __has_builtin(__builtin_amdgcn_global_load_async_to_lds_b128)
#define HAVE_ASYNC_LDS 1
#else
#define HAVE_ASYNC_LDS 0
#endif

static __device__ __forceinline__ void copy16_to_lds(const float* g, float* l) {
#if HAVE_ASYNC_LDS
  __builtin_amdgcn_global_load_async_to_lds_b128(
      (__attribute__((address_space(1))) v4i*)(uintptr_t)g,
      (__attribute__((address_space(3))) v4i*)(unsigned int)(uintptr_t)l,
      0, 0);
#else
  *(float4*)l = *(const float4*)g;
#endif
}

static __device__ __forceinline__ void wait_async_copies() {
#if HAVE_ASYNC_LDS
#if __has_builtin(__builtin_amdgcn_s_wait_asynccnt)
  __builtin_amdgcn_s_wait_asynccnt(0);
#else
  asm volatile("s_wait_asynccnt 0x0" ::: "memory");
#endif
#endif
}

// ---------------------------------------------------------------------------
// Pack weight (N_out x K row-major, K a power of two) into K-pair interleaved
// B-operand layout: Bp[(k>>1)*2N + 2n + (k&1)] = W[n*K + k]
// ---------------------------------------------------------------------------
__global__ void __launch_bounds__(256)
pack_weight(const float* __restrict__ w, float* __restrict__ bp, int N, int kshift) {
  const int K = 1 << kshift;
  int i = blockIdx.x * 256 + threadIdx.x;
  if (i >= N * K) return;
  int n = i >> kshift, k = i & (K - 1);
  bp[(size_t)(k >> 1) * (2 * N) + 2 * n + (k & 1)] = w[i];
}

// ---------------------------------------------------------------------------
// Generic GEMM: out[m][n] = act( sum_k A[m][k] * W[n][k] + bias[n] )
// Block = 256 threads = 8 waves; wave -> one 16x16 tile; grid(N/64, M/32).
// act: 0 = identity, 1 = relu, 2 = softplus
// ---------------------------------------------------------------------------
__global__ void __launch_bounds__(256)
gemm_bias_act(const float* __restrict__ A, int lda,
              const float* __restrict__ Bp, int N,
              const float* __restrict__ bias,
              float* __restrict__ out, int ldo,
              int K, int act)
{
  const int lane = threadIdx.x & 31;
  const int wave = threadIdx.x >> 5;
  const int half = lane >> 4, l16 = lane & 15;
  const int m0 = blockIdx.y * 32 + (wave >> 2) * 16;
  const int n0 = blockIdx.x * 64 + (wave & 3) * 16;
  const float* arow = A + (size_t)(m0 + l16) * lda + half * 2;
  const int col = n0 + l16;
  const float* brow = Bp + (size_t)half * (2 * N) + 2 * col;

  v8f acc = {};
  for (int k = 0; k < K; k += 4) {
    // A 16x4 layout: lanes 0-15 hold K=k..k+1, lanes 16-31 hold K=k+2..k+3
    v2f a = *(const v2f*)(arow + k);
    // B 4x16 layout mirrors it: VGPR0=K(k+2*half), VGPR1=K(k+2*half+1)
    v2f b = *(const v2f*)(brow + (size_t)(k >> 1) * (2 * N));
    acc = wmma4(a, b, acc);
  }

  const float bv = bias[col];
#pragma unroll
  for (int r = 0; r < 8; ++r) {
    float v = acc[r] + bv;
    if (act == 1)      v = fmaxf(v, 0.0f);
    else if (act == 2) v = fast_softplus(v);
    out[(size_t)(m0 + half * 8 + r) * ldo + col] = v;
  }
}

// ---------------------------------------------------------------------------
// Fused ODE step:
//   f = tanh(g2 @ f_w2^T + f_b2)  (B x H x D, never materialized)
//   y_out = y_in + dt * einsum('bhd,bd->bh', f, deriv)
// Grid: (H/8, B/32); block 256 = 8 waves; wave w handles h = h0+w, 32 rows.
// ---------------------------------------------------------------------------
__global__ void __launch_bounds__(256)
cde_step(const float* __restrict__ g2,   // B x W
         const float* __restrict__ w2p,  // packed W x HD
         const float* __restrict__ fb2,  // HD
         const float* __restrict__ y_in, // B x H
         float* __restrict__ y_out,      // B x H
         const float* __restrict__ ts,
         const float* __restrict__ cb,
         const float* __restrict__ cc,
         const float* __restrict__ cd,
         int step)
{
  __shared__ float sA[32 * 260];  // g2 slab, padded stride (conflict-free b64 reads)
  __shared__ float sD[32 * 68];   // deriv slab, padded

  const int tid = threadIdx.x;
  const int m0 = blockIdx.y * 32;
  const int h  = blockIdx.x * 8 + (tid >> 5);

  const float dt = ts[1] - ts[0];
  const float t  = ts[0] + (float)step * dt;
  // searchsorted(ts, t, 'right') - 1, clipped to [0, T-2]
  int lo = 0, hi = TT;
  while (lo < hi) { int mid = (lo + hi) >> 1; if (ts[mid] <= t) lo = mid + 1; else hi = mid; }
  int idx = lo - 1;
  idx = idx < 0 ? 0 : (idx > TT - 2 ? TT - 2 : idx);
  const float frac = t - ts[idx];

  // stage g2 slab asynchronously: 32 rows x 256 cols, one b128 per thread per iter
#pragma unroll
  for (int i = tid; i < 2048; i += 256) {
    int row = i >> 6;
    int c4  = (i & 63) * 4;
    copy16_to_lds(g2 + (size_t)(m0 + row) * WW + c4, sA + row * 260 + c4);
  }
  // deriv[b][d] = cb + 2*cc*frac + 3*cd*frac^2  for 32 rows x 64 d (computed, not copied)
  const float fr2 = frac * frac;
  for (int i = tid; i < 32 * 64; i += 256) {
    int row = i >> 6, d = i & 63;
    size_t off = ((size_t)(m0 + row) * (TT - 1) + idx) * DD + d;
    sD[row * 68 + d] = cb[off] + 2.0f * cc[off] * frac + 3.0f * cd[off] * fr2;
  }
  wait_async_copies();
  __syncthreads();

  const int lane = tid & 31, half = lane >> 4, l16 = lane & 15;

  v8f acc[2][4] = {};
  float fb[4];
  int colb[4];
#pragma unroll
  for (int j = 0; j < 4; ++j) {
    fb[j]   = fb2[h * 64 + j * 16 + l16];
    colb[j] = 2 * (h * 64 + j * 16 + l16);
  }

  for (int k = 0; k < WW; k += 4) {
    v2f a0 = *(const v2f*)(sA + l16 * 260 + k + half * 2);
    v2f a1 = *(const v2f*)(sA + (16 + l16) * 260 + k + half * 2);
    const float* bbase = w2p + ((size_t)(k >> 1) + half) * (2 * HD);
    if (k + 4 < WW)
      __builtin_prefetch(bbase + (size_t)4 * HD + 2 * (h * 64), 0, 0);
#pragma unroll
    for (int j = 0; j < 4; ++j) {
      v2f b = *(const v2f*)(bbase + colb[j]);
      acc[0][j] = wmma4(a0, b, acc[0][j]);
      acc[1][j] = wmma4(a1, b, acc[1][j]);
    }
  }

  // tanh, multiply by deriv, reduce over d (in-lane over j, cross-lane over l16)
#pragma unroll
  for (int mt = 0; mt < 2; ++mt) {
#pragma unroll
    for (int r = 0; r < 8; ++r) {
      const int rl = mt * 16 + half * 8 + r;  // local batch row
      float s = 0.0f;
#pragma unroll
      for (int j = 0; j < 4; ++j) {
        float v = fast_tanh(acc[mt][j][r] + fb[j]);
        s += v * sD[rl * 68 + j * 16 + l16];
      }
      s += __shfl_xor(s, 1, 32);
      s += __shfl_xor(s, 2, 32);
      s += __shfl_xor(s, 4, 32);
      s += __shfl_xor(s, 8, 32);
      if (l16 == 0) {
        const int row = m0 + rl;
        y_out[(size_t)row * HH + h] = y_in[(size_t)row * HH + h] + dt * s;
      }
    }
  }
}

// ---------------------------------------------------------------------------
// out[b] = sigmoid( dot(y[b,:], l_w) + l_b )
// ---------------------------------------------------------------------------
__global__ void __launch_bounds__(256)
final_sigmoid(const float* __restrict__ y, const float* __restrict__ lw,
              const float* __restrict__ lb, float* __restrict__ out)
{
  int b = threadIdx.x;  // grid = 1 block of 256
  float s = lb[0];
  for (int h = 0; h < HH; ++h) s += y[(size_t)b * HH + h] * lw[h];
  out[b] = 1.0f / (1.0f + __expf(-s));
}

extern "C" void kernel_launch(void* const* d_in, const int* in_sizes, int n_in,
                              void* d_out, int out_size, void* d_ws, size_t ws_size,
                              hipStream_t stream)
{
  const float* ts   = (const float*)d_in[0];
  const float* cd_  = (const float*)d_in[1];
  const float* cc_  = (const float*)d_in[2];
  const float* cb_  = (const float*)d_in[3];
  const float* ca_  = (const float*)d_in[4];
  const float* i_w0 = (const float*)d_in[5];
  const float* i_b0 = (const float*)d_in[6];
  const float* i_w1 = (const float*)d_in[7];
  const float* i_b1 = (const float*)d_in[8];
  const float* i_w2 = (const float*)d_in[9];
  const float* i_b2 = (const float*)d_in[10];
  const float* f_w0 = (const float*)d_in[11];
  const float* f_b0 = (const float*)d_in[12];
  const float* f_w1 = (const float*)d_in[13];
  const float* f_b1 = (const float*)d_in[14];
  const float* f_w2 = (const float*)d_in[15];
  const float* f_b2 = (const float*)d_in[16];
  const float* l_w  = (const float*)d_in[17];
  const float* l_b  = (const float*)d_in[18];

  float* ws = (float*)d_ws;
  float* iw0p = ws;               ws += (size_t)WW * DD;
  float* iw1p = ws;               ws += (size_t)WW * WW;
  float* iw2p = ws;               ws += (size_t)HH * WW;
  float* fw0p = ws;               ws += (size_t)WW * HH;
  float* fw1p = ws;               ws += (size_t)WW * WW;
  float* fw2p = ws;               ws += (size_t)HD * WW;
  float* g1   = ws;               ws += (size_t)BB * WW;
  float* g2   = ws;               ws += (size_t)BB * WW;
  float* yA   = ws;               ws += (size_t)BB * HH;
  float* yB   = ws;               ws += (size_t)BB * HH;

  // One-time weight packing (re-done every call; deterministic). K is pow2 -> pass shift.
  pack_weight<<<(WW * DD + 255) / 256, 256, 0, stream>>>(i_w0, iw0p, WW, 6);
  pack_weight<<<(WW * WW + 255) / 256, 256, 0, stream>>>(i_w1, iw1p, WW, 8);
  pack_weight<<<(HH * WW + 255) / 256, 256, 0, stream>>>(i_w2, iw2p, HH, 8);
  pack_weight<<<(WW * HH + 255) / 256, 256, 0, stream>>>(f_w0, fw0p, WW, 7);
  pack_weight<<<(WW * WW + 255) / 256, 256, 0, stream>>>(f_w1, fw1p, WW, 8);
  pack_weight<<<(HD * WW + 255) / 256, 256, 0, stream>>>(f_w2, fw2p, HD, 8);

  // Init MLP: x0 = coeffs_a[:,0,:]  (lda = (T-1)*D picks the idx-0 slice)
  gemm_bias_act<<<dim3(WW / 64, BB / 32), 256, 0, stream>>>(
      ca_, (TT - 1) * DD, iw0p, WW, i_b0, g1, WW, DD, 1);
  gemm_bias_act<<<dim3(WW / 64, BB / 32), 256, 0, stream>>>(
      g1, WW, iw1p, WW, i_b1, g2, WW, WW, 1);
  gemm_bias_act<<<dim3(HH / 64, BB / 32), 256, 0, stream>>>(
      g2, WW, iw2p, HH, i_b2, yA, HH, WW, 0);

  float* y_cur = yA;
  float* y_nxt = yB;
  for (int step = 0; step < TT; ++step) {
    gemm_bias_act<<<dim3(WW / 64, BB / 32), 256, 0, stream>>>(
        y_cur, HH, fw0p, WW, f_b0, g1, WW, HH, 2);
    gemm_bias_act<<<dim3(WW / 64, BB / 32), 256, 0, stream>>>(
        g1, WW, fw1p, WW, f_b1, g2, WW, WW, 2);
    cde_step<<<dim3(HH / 8, BB / 32), 256, 0, stream>>>(
        g2, fw2p, f_b2, y_cur, y_nxt, ts, cb_, cc_, cd_, step);
    float* tmp = y_cur; y_cur = y_nxt; y_nxt = tmp;
  }

  final_sigmoid<<<1, 256, 0, stream>>>(y_cur, l_w, l_b, (float*)d_out);
}